// LSTMModel_37254546326201
// MI455X (gfx1250) — compile-verified
//
#include <hip/hip_runtime.h>

typedef __attribute__((ext_vector_type(2))) float v2f;
typedef __attribute__((ext_vector_type(8))) float v8f;

#define HSZ   15
#define GSZ   60      // 4*H
#define DIN   5
#define TSEQ  512
#define BTOT  4096
#define ONUM  25

__device__ __forceinline__ float sigm(float x) {
    return 1.0f / (1.0f + __expf(-x));
}
__device__ __forceinline__ float tanh_fast(float x) {
    return 2.0f / (1.0f + __expf(-2.0f * x)) - 1.0f;
}

// Merged weight matrix for layer 0 (K=24, N=64):
//   rows 0..14  = w_hh0^T, row 15 = 0,
//   rows 16..20 = w_ih0^T,
//   row 21      = bias (b_ih0+b_hh0)  [paired with constant-1.0 in A],
//   rows 22..23 = 0.
__device__ __forceinline__ float w0val(const float* w_ih0, const float* w_hh0,
                                       const float* b_ih0, const float* b_hh0,
                                       int k, int g) {
    if (g >= GSZ) return 0.0f;
    if (k < HSZ) return w_hh0[g * HSZ + k];
    if (k >= 16 && k < 16 + DIN) return w_ih0[g * DIN + (k - 16)];
    if (k == 21) return b_ih0[g] + b_hh0[g];
    return 0.0f;
}
// Merged weight matrix for layer 1 (K=32, N=64):
//   rows 0..14  = w_hh1^T, row 15 = 0,
//   rows 16..30 = w_ih1^T,
//   row 31      = bias (b_ih1+b_hh1)  [paired with constant-1.0 in st1[.][31]].
__device__ __forceinline__ float w1val(const float* w_ih1, const float* w_hh1,
                                       const float* b_ih1, const float* b_hh1,
                                       int k, int g) {
    if (g >= GSZ) return 0.0f;
    if (k < HSZ) return w_hh1[g * HSZ + k];
    if (k >= 16 && k < 16 + HSZ) return w_ih1[g * HSZ + (k - 16)];
    if (k == 31) return b_ih1[g] + b_hh1[g];
    return 0.0f;
}

__global__ __launch_bounds__(32)
void lstm2_wmma_kernel(const float* __restrict__ x,
                       const float* __restrict__ w_ih0, const float* __restrict__ w_hh0,
                       const float* __restrict__ b_ih0, const float* __restrict__ b_hh0,
                       const float* __restrict__ w_ih1, const float* __restrict__ w_hh1,
                       const float* __restrict__ b_ih1, const float* __restrict__ b_hh1,
                       const float* __restrict__ w_lin, const float* __restrict__ b_lin,
                       float* __restrict__ out)
{
    // Per-wave LDS staging (one wave per block).
    __shared__ float st0[16][16];   // [batch][k] : h0(15), pad           (A source, layer 0)
    __shared__ float st1[16][32];   // [batch][k] : h1(15),pad, h0(15),1  (A source, layer 1)
    __shared__ float gt[64][16];    // [gate][batch] : transposed gate pre-activations

    const int lane  = threadIdx.x;   // 0..31
    const int bl    = lane & 15;
    const int hi    = lane >> 4;     // lane half
    const int bbase = blockIdx.x * 16;

    // ---- Pack weight B-operands into registers (WMMA B layout, K=4 chunks) ----
    // chunk j, tile nt: v.x = B[4j+2*hi][g], v.y = B[4j+2*hi+1][g], g = nt*16+bl
    v2f B0[6][4];
    v2f B1[8][4];
#pragma unroll
    for (int j = 0; j < 6; ++j) {
#pragma unroll
        for (int nt = 0; nt < 4; ++nt) {
            const int k0 = 4 * j + 2 * hi;
            const int g  = nt * 16 + bl;
            B0[j][nt].x = w0val(w_ih0, w_hh0, b_ih0, b_hh0, k0,     g);
            B0[j][nt].y = w0val(w_ih0, w_hh0, b_ih0, b_hh0, k0 + 1, g);
        }
    }
#pragma unroll
    for (int j = 0; j < 8; ++j) {
#pragma unroll
        for (int nt = 0; nt < 4; ++nt) {
            const int k0 = 4 * j + 2 * hi;
            const int g  = nt * 16 + bl;
            B1[j][nt].x = w1val(w_ih1, w_hh1, b_ih1, b_hh1, k0,     g);
            B1[j][nt].y = w1val(w_ih1, w_hh1, b_ih1, b_hh1, k0 + 1, g);
        }
    }

    // Zero hidden-state staging + cell state; plant constant-1.0 at st1[.][31].
    for (int i = lane; i < 16 * 16; i += 32) st0[i / 16][i % 16] = 0.0f;
    for (int i = lane; i < 16 * 32; i += 32) st1[i / 32][i % 32] = 0.0f;
    st1[bl][31] = 1.0f;
    float c0r[8], c1r[8];
#pragma unroll
    for (int u = 0; u < 8; ++u) { c0r[u] = 0.0f; c1r[u] = 0.0f; }

    const v8f zacc = {0.f, 0.f, 0.f, 0.f, 0.f, 0.f, 0.f, 0.f};

    // Software-pipelined x load: fetch x(t) before the loop, x(t+1) inside it.
    // Lane L<16 fetches x[b][0],x[b][1],x[b][4]; lane L>=16 fetches x[b][2],x[b][3],(x[b][4] unused).
    const float* xptr = x + ((size_t)0 * BTOT + (bbase + bl)) * DIN;
    const int xoff = hi ? 2 : 0;
    float xr0 = xptr[xoff], xr1 = xptr[xoff + 1], xr2 = xptr[4];

    __syncthreads();

    for (int t = 0; t < TSEQ; ++t) {
        // ================= layer 0 =================
        // A chunks: k 0..15 from st0 (h0 + pad), k 16..23 from x | 1.0 | pad.
        v2f a[8];
#pragma unroll
        for (int j = 0; j < 4; ++j) {
            a[j].x = st0[bl][4 * j + 2 * hi];
            a[j].y = st0[bl][4 * j + 2 * hi + 1];
        }
        a[4].x = xr0;                  // k=16+2*hi
        a[4].y = xr1;                  // k=17+2*hi
        a[5].x = hi ? 0.0f : xr2;      // k=20 : x4   (k=22 : 0)
        a[5].y = hi ? 0.0f : 1.0f;     // k=21 : 1.0  (k=23 : 0)

        // Prefetch x(t+1) — latency hidden behind the full two-layer step.
        if (t + 1 < TSEQ) {
            xptr += BTOT * DIN;
            xr0 = xptr[xoff]; xr1 = xptr[xoff + 1]; xr2 = xptr[4];
        }

        v8f acc[4];
#pragma unroll
        for (int nt = 0; nt < 4; ++nt) {
            acc[nt] = __builtin_amdgcn_wmma_f32_16x16x4_f32(
                false, a[0], false, B0[0][nt], (short)0, zacc, false, false);
#pragma unroll
            for (int j = 1; j < 6; ++j) {
                acc[nt] = __builtin_amdgcn_wmma_f32_16x16x4_f32(
                    false, a[j], false, B0[j][nt], (short)0, acc[nt], false, false);
            }
        }
        // D layout -> gt[gate][batch]; value acc[nt][r] is (m = r+8*hi, g = nt*16+bl)
#pragma unroll
        for (int nt = 0; nt < 4; ++nt) {
#pragma unroll
            for (int r = 0; r < 8; ++r) gt[nt * 16 + bl][8 * hi + r] = acc[nt][r];
        }
        __syncthreads();

        // Elementwise cell update, layer 0. Lane owns batch=bl, hids 8*hi..8*hi+7.
#pragma unroll
        for (int u = 0; u < 8; ++u) {
            const int hid = 8 * hi + u;
            if (hid < HSZ) {
                const float iv = sigm(gt[hid][bl]);
                const float fv = sigm(gt[HSZ + hid][bl]);
                const float gv = tanh_fast(gt[2 * HSZ + hid][bl]);
                const float ov = sigm(gt[3 * HSZ + hid][bl]);
                c0r[u] = fv * c0r[u] + iv * gv;
                const float h = ov * tanh_fast(c0r[u]);
                st0[bl][hid]      = h;   // recurrent input, layer 0
                st1[bl][16 + hid] = h;   // feed-forward input, layer 1
            }
        }
        __syncthreads();

        // ================= layer 1 =================
#pragma unroll
        for (int j = 0; j < 8; ++j) {
            a[j].x = st1[bl][4 * j + 2 * hi];
            a[j].y = st1[bl][4 * j + 2 * hi + 1];
        }
#pragma unroll
        for (int nt = 0; nt < 4; ++nt) {
            acc[nt] = __builtin_amdgcn_wmma_f32_16x16x4_f32(
                false, a[0], false, B1[0][nt], (short)0, zacc, false, false);
#pragma unroll
            for (int j = 1; j < 8; ++j) {
                acc[nt] = __builtin_amdgcn_wmma_f32_16x16x4_f32(
                    false, a[j], false, B1[j][nt], (short)0, acc[nt], false, false);
            }
        }
#pragma unroll
        for (int nt = 0; nt < 4; ++nt) {
#pragma unroll
            for (int r = 0; r < 8; ++r) gt[nt * 16 + bl][8 * hi + r] = acc[nt][r];
        }
        __syncthreads();

#pragma unroll
        for (int u = 0; u < 8; ++u) {
            const int hid = 8 * hi + u;
            if (hid < HSZ) {
                const float iv = sigm(gt[hid][bl]);
                const float fv = sigm(gt[HSZ + hid][bl]);
                const float gv = tanh_fast(gt[2 * HSZ + hid][bl]);
                const float ov = sigm(gt[3 * HSZ + hid][bl]);
                c1r[u] = fv * c1r[u] + iv * gv;
                const float h = ov * tanh_fast(c1r[u]);
                st1[bl][hid] = h;        // recurrent input, layer 1
            }
        }
        __syncthreads();
    }

    // ---- Final linear: out[b][o] = h1_last[b] . w_lin[o] + b_lin[o] ----
    for (int idx = lane; idx < 16 * ONUM; idx += 32) {
        const int b = idx / ONUM;
        const int o = idx % ONUM;
        float s = b_lin[o];
#pragma unroll
        for (int k = 0; k < HSZ; ++k) s += st1[b][k] * w_lin[o * HSZ + k];
        out[(size_t)(bbase + b) * ONUM + o] = s;
    }
}

extern "C" void kernel_launch(void* const* d_in, const int* in_sizes, int n_in,
                              void* d_out, int out_size, void* d_ws, size_t ws_size,
                              hipStream_t stream) {
    const float* x     = (const float*)d_in[0];
    const float* w_ih0 = (const float*)d_in[1];
    const float* w_hh0 = (const float*)d_in[2];
    const float* b_ih0 = (const float*)d_in[3];
    const float* b_hh0 = (const float*)d_in[4];
    const float* w_ih1 = (const float*)d_in[5];
    const float* w_hh1 = (const float*)d_in[6];
    const float* b_ih1 = (const float*)d_in[7];
    const float* b_hh1 = (const float*)d_in[8];
    const float* w_lin = (const float*)d_in[9];
    const float* b_lin = (const float*)d_in[10];
    float* out = (float*)d_out;

    dim3 grid(BTOT / 16);   // 256 waves, one 16-row batch tile each
    dim3 block(32);         // one wave32 per workgroup
    lstm2_wmma_kernel<<<grid, block, 0, stream>>>(
        x, w_ih0, w_hh0, b_ih0, b_hh0,
        w_ih1, w_hh1, b_ih1, b_hh1, w_lin, b_lin, out);
}